// ATOM_RNN_10806137716880
// MI455X (gfx1250) — compile-verified
//
#include <hip/hip_runtime.h>
#include <hip/hip_bf16.h>
#include <cstdint>

// ---------------- problem constants ----------------
#define BTOT   65536
#define OBSD   85
#define FEAT   8
#define NL     5
#define NT     10
#define INF    64
#define HID    64
#define NACT   5

#define SPW    16      // samples per wave (WMMA M-tile)
#define NWAVES 2       // waves per block
// padded LDS leading dims (keep 16 rows on distinct banks: ld % 64 != 0)
#define LD_OBS 92      // 85 obs + 5 obs_out + 2 zero pad (K=92 for fc1)
#define LD_68  68
#define LD_HP  164     // 5*32 = 160 -> 164
#define LD_HP2 84      // 5*16 = 80  -> 84
#define LD_G   196     // 192 -> 196

// ---- packed-weight workspace layout (float offsets, all even -> 8B aligned) ----
#define OFF_U    0        // 128 floats: u0|u1 (plain, for fallback VALU fold)
#define OFF_IN0  128      // 512  (K=8,  N=64)
#define OFF_IN1  640      // 512
#define OFF_IN2  1152     // 512
#define OFF_O1   1664     // 384  (K=10 pad 12, N=32)
#define OFF_O2   2048     // 512  (K=32, N=16)
#define OFF_FC1  2560     // 5888 (K=90 pad 92, N=64)
#define OFF_IH   8448     // 12288 (K=64, N=192, transposed from (192,64))
#define OFF_HH   20736    // 12288
#define OFF_UP   33024    // 1024 (U-matrix 64x16: cols 0/1 = u0/u1, frag-packed)
#define WS_FLOATS 34048   // ~133 KB

typedef float v2f __attribute__((ext_vector_type(2)));
typedef float v8f __attribute__((ext_vector_type(8)));

__device__ __forceinline__ v8f wmma4(v2f a, v2f b, v8f c) {
  // V_WMMA_F32_16X16X4_F32 : D = A(16x4,f32) * B(4x16,f32) + C(16x16,f32)
  return __builtin_amdgcn_wmma_f32_16x16x4_f32(
      false, a, false, b, (short)0, c, false, false);
}

// one coalesced global_load_b64 per lane = full 4x16 B fragment per wave
__device__ __forceinline__ v2f bfragP(const float* __restrict__ base,
                                      int kt, int nt, int ntiles, int lane) {
  return ((const v2f*)base)[(kt * ntiles + nt) * 32 + lane];
}

// ALPHA=0.01 < 1  =>  leaky(x) == max(x, 0.01*x) : 2 VALU ops, no VCC chain
__device__ __forceinline__ float leaky(float x) { return fmaxf(x, 0.01f * x); }
__device__ __forceinline__ float sigm(float x)  { return 1.f / (1.f + __expf(-x)); }

// ---- precompute u0 = W @ a[:64], u1 = W @ a[64:]; also frag-packed U(64x16) ----
__global__ void atom_rnn_prep_u(const float* __restrict__ W,
                                const float* __restrict__ a,
                                float* __restrict__ u,
                                float* __restrict__ up, int do_pack) {
  __shared__ float su[2 * INF];
  int i = threadIdx.x;           // 64 threads
  float s0 = 0.f, s1 = 0.f;
  for (int j = 0; j < INF; ++j) {
    float w = W[i * INF + j];
    s0 += w * a[j];
    s1 += w * a[INF + j];
  }
  u[i] = s0;
  u[INF + i] = s1;
  su[i] = s0;
  su[INF + i] = s1;
  __syncthreads();
  if (do_pack) {
    // U[k][n] = n==0 ? u0[k] : n==1 ? u1[k] : 0 ; fragment order, 16 kt x 1 nt
    for (int e = i; e < 1024; e += 64) {
      int p = e & 1, lane = (e >> 1) & 31, kt = e >> 6;
      int k = (kt << 2) + ((lane >> 4) << 1) + p;
      int n = lane & 15;
      up[e] = (n < 2) ? su[n * INF + k] : 0.f;
    }
  }
}

// ---------------- pack a weight matrix into WMMA B-fragment order ----------------
// dst[((kt*ntiles + nt)*32 + lane)*2 + p] = B[kt*4 + (lane>=16)*2 + p][nt*16 + lane%16]
// B[k][n] = trans ? src[n*K + k] : src[k*N + n];  k >= Kvalid -> 0 (K padding baked in)
__global__ void atom_rnn_pack(const float* __restrict__ src, float* __restrict__ dst,
                              int Kvalid, int Kpad, int N, int trans) {
  int ntiles = N >> 4;
  int total = (Kpad >> 2) * ntiles * 64;
  for (int i = blockIdx.x * blockDim.x + threadIdx.x; i < total;
       i += gridDim.x * blockDim.x) {
    int p    = i & 1;
    int lane = (i >> 1) & 31;
    int tile = i >> 6;
    int nt = tile % ntiles, kt = tile / ntiles;
    int k = (kt << 2) + ((lane >> 4) << 1) + p;
    int n = (nt << 4) + (lane & 15);
    float v = 0.f;
    if (k < Kvalid) v = trans ? src[n * Kvalid + k] : src[k * N + n];
    dst[i] = v;
  }
}

// ---------------- fused forward: 16 samples per wave ----------------
template <bool PACKED>
__global__ __launch_bounds__(NWAVES * 32) void atom_rnn_fused(
    const float* __restrict__ obs, const float* __restrict__ hid,
    const float* __restrict__ w_in0, const float* __restrict__ b_in0,
    const float* __restrict__ w_in1, const float* __restrict__ b_in1,
    const float* __restrict__ w_in2, const float* __restrict__ b_in2,
    const float* __restrict__ wsP,                       // workspace (u + packed weights)
    const float* __restrict__ w_o1, const float* __restrict__ b_o1,
    const float* __restrict__ w_o2, const float* __restrict__ b_o2,
    const float* __restrict__ w_o3, const float* __restrict__ b_o3,
    const float* __restrict__ w_fc1, const float* __restrict__ b_fc1,
    const float* __restrict__ w_ih,  const float* __restrict__ w_hh,
    const float* __restrict__ b_ih,  const float* __restrict__ b_hh,
    const float* __restrict__ w_fc2, const float* __restrict__ b_fc2,
    float* __restrict__ outQ, float* __restrict__ outH) {

  __shared__ __align__(16) float s_obs [NWAVES][SPW * LD_OBS];
  __shared__ __align__(16) float s_hst [NWAVES][SPW * LD_68];
  __shared__ __align__(16) float s_wh  [NWAVES][2 * SPW * NT];
  __shared__ __align__(16) float s_bufA[NWAVES][4224];  // stage tile -> hp -> x|gi
  __shared__ __align__(16) float s_bufB[NWAVES][3136];  // att -> hp2 -> gh

  const int wv   = threadIdx.x >> 5;
  const int lane = threadIdx.x & 31;
  const int s0   = (blockIdx.x * NWAVES + wv) * SPW;

  float* obsW = s_obs[wv];
  float* hstW = s_hst[wv];
  float* whW  = s_wh[wv];
  float* bufA = s_bufA[wv];
  float* bufB = s_bufB[wv];

  const float* uvec = wsP + OFF_U;

  // per-lane WMMA fragment coordinates
  const int lm = lane & 15;              // A row / B,C,D column
  const int kb = (lane >> 4) << 1;       // K sub-offset: lanes 16-31 hold K+2,K+3
  const int mb = (lane >> 4) << 3;       // C/D row base: lanes 16-31 hold M=8..15

  // ---------------- stage 0: load obs + hidden into LDS (no int div) -----------
  for (int s = 0; s < SPW; ++s) {
    const float* src = obs + (size_t)(s0 + s) * OBSD;
    for (int c = lane; c < OBSD; c += 32) obsW[s * LD_OBS + c] = src[c];
  }
  if (lane < 7) {                        // zero cols 85..91
    for (int s = 0; s < SPW; ++s) obsW[s * LD_OBS + OBSD + lane] = 0.f;
  }
  for (int idx = lane; idx < SPW * HID; idx += 32) {
    int s = idx >> 6, c = idx & 63;
    hstW[s * LD_68 + c] = hid[(size_t)(s0 + s) * HID + c];
  }

  // ---------------- stage A: input nets + Wh1/Wh2 fold -------------------------
  // h_mix row t = leaky(obs_in[:,t,:] @ w_sel(t) + b_sel(t)); obs_in[t][f]=o[(4+8t+f)%80]
  // Wh eliminated: Wh1[:,t] = Hd_t @ u0, Wh2[:,t] = Hd_t @ u1 (WMMA with packed U).
  float* stageW = bufA;
  for (int t = 0; t < NT; ++t) {
    const float* wt  = (t < NL) ? w_in0 : (t < NT - 1) ? w_in1 : w_in2;
    const float* bt  = (t < NL) ? b_in0 : (t < NT - 1) ? b_in1 : b_in2;
    const float* wtp = wsP + ((t < NL) ? OFF_IN0 : (t < NT - 1) ? OFF_IN1 : OFF_IN2);
    for (int n0 = 0; n0 < INF; n0 += 16) {
      v8f acc = {0.f, 0.f, 0.f, 0.f, 0.f, 0.f, 0.f, 0.f};
      #pragma unroll
      for (int k0 = 0; k0 < FEAT; k0 += 4) {
        int f0 = k0 + kb;
        // rotated column; c0 is even and < 160, and the pair never straddles the
        // wrap point, so %80 is a single conditional subtract (no div chain)
        int c0 = 4 + 8 * t + f0;
        c0 = (c0 >= 80) ? c0 - 80 : c0;
        v2f a, b;
        a.x = obsW[lm * LD_OBS + c0];
        a.y = obsW[lm * LD_OBS + c0 + 1];
        if (PACKED) {
          b = bfragP(wtp, k0 >> 2, n0 >> 4, 4, lane);
        } else {
          b.x = wt[f0 * INF + n0 + lm];
          b.y = wt[(f0 + 1) * INF + n0 + lm];
        }
        acc = wmma4(a, b, acc);
      }
      float bb = bt[n0 + lm];
      #pragma unroll
      for (int v = 0; v < 8; ++v)
        stageW[(v + mb) * LD_68 + n0 + lm] = leaky(acc[v] + bb);
    }
    if (PACKED) {   // fold on the matrix pipe: Wh(16x16) = Hd(16x64) @ U(64x16)
      v8f acc = {0.f, 0.f, 0.f, 0.f, 0.f, 0.f, 0.f, 0.f};
      #pragma unroll 4
      for (int k0 = 0; k0 < INF; k0 += 4) {
        v2f a;
        a.x = stageW[lm * LD_68 + k0 + kb];
        a.y = stageW[lm * LD_68 + k0 + kb + 1];
        acc = wmma4(a, bfragP(wsP + OFF_UP, k0 >> 2, 0, 1, lane), acc);
      }
      if (lm < 2) {  // D cols 0/1 = Wh1/Wh2 rows
        #pragma unroll
        for (int v = 0; v < 8; ++v)
          whW[lm * (SPW * NT) + (v + mb) * NT + t] = acc[v];
      }
    } else {        // fallback: VALU dots (lanes 0-15 -> u0, 16-31 -> u1)
      int s = lm, sel = lane >> 4;
      const float* u = uvec + sel * INF;
      float acc = 0.f;
      for (int k = 0; k < INF; ++k) acc += stageW[s * LD_68 + k] * u[k];
      whW[sel * (SPW * NT) + s * NT + t] = acc;
    }
  }

  // ---------------- stage B1: attention coefficients (lanes 0-15) --------------
  float* attW = bufB;                        // [s][r*12 + c], ld 68, c 10..11 zero
  if (lane < SPW) {
    int s = lane;
    float wh1[NT], wh2[NT];
    for (int t = 0; t < NT; ++t) {
      wh1[t] = whW[s * NT + t];
      wh2[t] = whW[SPW * NT + s * NT + t];
    }
    for (int r = 0; r < NL; ++r) {           // att0 rows: softmax_j e[r][5+j]
      float ev[NL], mx = -1e30f;
      for (int j = 0; j < NL; ++j) { ev[j] = leaky(wh1[r] + wh2[NL + j]); mx = fmaxf(mx, ev[j]); }
      float sum = 0.f;
      for (int j = 0; j < NL; ++j) { ev[j] = __expf(ev[j] - mx); sum += ev[j]; }
      float inv = 1.f / sum;
      for (int j = 0; j < NL; ++j) attW[s * LD_68 + r * 12 + j] = ev[j] * inv;
    }
    for (int k = 0; k < NL; ++k) {           // att1^T cols: softmax_r e[5+k][r]
      float ev[NL], mx = -1e30f;
      for (int r = 0; r < NL; ++r) { ev[r] = leaky(wh1[NL + k] + wh2[r]); mx = fmaxf(mx, ev[r]); }
      float sum = 0.f;
      for (int r = 0; r < NL; ++r) { ev[r] = __expf(ev[r] - mx); sum += ev[r]; }
      float inv = 1.f / sum;
      for (int r = 0; r < NL; ++r) attW[s * LD_68 + r * 12 + NL + k] = ev[r] * inv;
    }
    for (int r = 0; r < NL; ++r) {
      attW[s * LD_68 + r * 12 + 10] = 0.f;
      attW[s * LD_68 + r * 12 + 11] = 0.f;
    }
  }

  // ---------------- stage B2: hp1 = leaky(att @ w_o1 + b_o1)  (K=12, N=32) -----
  float* hpW = bufA;                         // [s][r*32 + c], ld 164
  for (int r = 0; r < NL; ++r) {
    for (int n0 = 0; n0 < 32; n0 += 16) {
      v8f acc = {0.f, 0.f, 0.f, 0.f, 0.f, 0.f, 0.f, 0.f};
      #pragma unroll
      for (int k0 = 0; k0 < 12; k0 += 4) {
        int k1 = k0 + kb;
        v2f a, b;
        a.x = attW[lm * LD_68 + r * 12 + k1];
        a.y = attW[lm * LD_68 + r * 12 + k1 + 1];
        if (PACKED) {
          b = bfragP(wsP + OFF_O1, k0 >> 2, n0 >> 4, 2, lane);
        } else {
          b.x = (k1     < NT) ? w_o1[k1 * 32 + n0 + lm]       : 0.f;
          b.y = (k1 + 1 < NT) ? w_o1[(k1 + 1) * 32 + n0 + lm] : 0.f;
        }
        acc = wmma4(a, b, acc);
      }
      float bb = b_o1[n0 + lm];
      #pragma unroll
      for (int v = 0; v < 8; ++v)
        hpW[(v + mb) * LD_HP + r * 32 + n0 + lm] = leaky(acc[v] + bb);
    }
  }

  // ---------------- stage B3: hp2 = leaky(hp1 @ w_o2 + b_o2)  (K=32, N=16) -----
  float* hp2W = bufB;                        // [s][r*16 + c], ld 84
  for (int r = 0; r < NL; ++r) {
    v8f acc = {0.f, 0.f, 0.f, 0.f, 0.f, 0.f, 0.f, 0.f};
    #pragma unroll
    for (int k0 = 0; k0 < 32; k0 += 4) {
      int k1 = k0 + kb;
      v2f a, b;
      a.x = hpW[lm * LD_HP + r * 32 + k1];
      a.y = hpW[lm * LD_HP + r * 32 + k1 + 1];
      if (PACKED) {
        b = bfragP(wsP + OFF_O2, k0 >> 2, 0, 1, lane);
      } else {
        b.x = w_o2[k1 * 16 + lm];
        b.y = w_o2[(k1 + 1) * 16 + lm];
      }
      acc = wmma4(a, b, acc);
    }
    float bb = b_o2[lm];
    #pragma unroll
    for (int v = 0; v < 8; ++v)
      hp2W[(v + mb) * LD_HP2 + r * 16 + lm] = leaky(acc[v] + bb);
  }

  // ---------------- stage B4: hp3 + softmax -> obs_out (lanes 0-15) ------------
  if (lane < SPW) {
    int s = lane;
    float hv[NL], mx = -1e30f;
    for (int r = 0; r < NL; ++r) {
      float acc = b_o3[0];
      for (int k = 0; k < 16; ++k) acc += hp2W[s * LD_HP2 + r * 16 + k] * w_o3[k];
      hv[r] = leaky(acc);
      mx = fmaxf(mx, hv[r]);
    }
    float sum = 0.f;
    for (int r = 0; r < NL; ++r) { hv[r] = __expf(hv[r] - mx); sum += hv[r]; }
    float inv = 1.f / sum;
    for (int r = 0; r < NL; ++r) obsW[s * LD_OBS + OBSD + r] = hv[r] * inv;
  }

  // ---------------- stage C1: x = relu(obs_comb @ w_fc1 + b_fc1) (K=92, N=64) --
  float* xW = bufA;                          // 16 x LD_68
  for (int n0 = 0; n0 < HID; n0 += 16) {
    v8f acc = {0.f, 0.f, 0.f, 0.f, 0.f, 0.f, 0.f, 0.f};
    for (int k0 = 0; k0 < LD_OBS; k0 += 4) {
      int k1 = k0 + kb;
      v2f a, b;
      a.x = obsW[lm * LD_OBS + k1];
      a.y = obsW[lm * LD_OBS + k1 + 1];
      if (PACKED) {
        b = bfragP(wsP + OFF_FC1, k0 >> 2, n0 >> 4, 4, lane);
      } else {
        b.x = (k1     < 90) ? w_fc1[k1 * HID + n0 + lm]       : 0.f;
        b.y = (k1 + 1 < 90) ? w_fc1[(k1 + 1) * HID + n0 + lm] : 0.f;
      }
      acc = wmma4(a, b, acc);
    }
    float bb = b_fc1[n0 + lm];
    #pragma unroll
    for (int v = 0; v < 8; ++v)
      xW[(v + mb) * LD_68 + n0 + lm] = fmaxf(acc[v] + bb, 0.f);
  }

  // ---------------- stage C2: gh = h_in @ w_hh^T + b_hh  (K=64, N=192) ---------
  float* ghW = bufB;                         // 16 x LD_G
  for (int n0 = 0; n0 < 3 * HID; n0 += 16) {
    v8f acc = {0.f, 0.f, 0.f, 0.f, 0.f, 0.f, 0.f, 0.f};
    #pragma unroll 4
    for (int k0 = 0; k0 < HID; k0 += 4) {
      int k1 = k0 + kb;
      v2f a, b;
      a.x = hstW[lm * LD_68 + k1];
      a.y = hstW[lm * LD_68 + k1 + 1];
      if (PACKED) {
        b = bfragP(wsP + OFF_HH, k0 >> 2, n0 >> 4, 12, lane);
      } else {
        b.x = w_hh[(n0 + lm) * HID + k1];
        b.y = w_hh[(n0 + lm) * HID + k1 + 1];
      }
      acc = wmma4(a, b, acc);
    }
    float bb = b_hh[n0 + lm];
    #pragma unroll
    for (int v = 0; v < 8; ++v)
      ghW[(v + mb) * LD_G + n0 + lm] = acc[v] + bb;
  }

  // ---------------- stage C3: gi = x @ w_ih^T + b_ih  (K=64, N=192) ------------
  float* giW = bufA + SPW * LD_68;
  for (int n0 = 0; n0 < 3 * HID; n0 += 16) {
    v8f acc = {0.f, 0.f, 0.f, 0.f, 0.f, 0.f, 0.f, 0.f};
    #pragma unroll 4
    for (int k0 = 0; k0 < HID; k0 += 4) {
      int k1 = k0 + kb;
      v2f a, b;
      a.x = xW[lm * LD_68 + k1];
      a.y = xW[lm * LD_68 + k1 + 1];
      if (PACKED) {
        b = bfragP(wsP + OFF_IH, k0 >> 2, n0 >> 4, 12, lane);
      } else {
        b.x = w_ih[(n0 + lm) * HID + k1];
        b.y = w_ih[(n0 + lm) * HID + k1 + 1];
      }
      acc = wmma4(a, b, acc);
    }
    float bb = b_ih[n0 + lm];
    #pragma unroll
    for (int v = 0; v < 8; ++v)
      giW[(v + mb) * LD_G + n0 + lm] = acc[v] + bb;
  }

  // ---------------- stage D: GRU cell + outputs --------------------------------
  float* hb = bufA;                          // reuse x region (x dead)
  for (int idx = lane; idx < SPW * HID; idx += 32) {
    int s = idx >> 6, c = idx & 63;
    float r  = sigm(giW[s * LD_G + c]            + ghW[s * LD_G + c]);
    float z  = sigm(giW[s * LD_G + HID + c]      + ghW[s * LD_G + HID + c]);
    float nn = tanhf(giW[s * LD_G + 2 * HID + c] + r * ghW[s * LD_G + 2 * HID + c]);
    float h  = (1.f - z) * nn + z * hstW[s * LD_68 + c];
    hb[s * LD_68 + c] = h;
    outH[(size_t)(s0 + s) * HID + c] = h;
  }
  for (int idx = lane; idx < SPW * NACT; idx += 32) {
    int s = idx / NACT, j = idx % NACT;
    float acc = b_fc2[j];
    for (int k = 0; k < HID; ++k) acc += hb[s * LD_68 + k] * w_fc2[k * NACT + j];
    outQ[(size_t)(s0 + s) * NACT + j] = acc;
  }
}

extern "C" void kernel_launch(void* const* d_in, const int* in_sizes, int n_in,
                              void* d_out, int out_size, void* d_ws, size_t ws_size,
                              hipStream_t stream) {
  const float* obs   = (const float*)d_in[0];
  const float* hid   = (const float*)d_in[1];
  const float* w_in0 = (const float*)d_in[2];
  const float* b_in0 = (const float*)d_in[3];
  const float* w_in1 = (const float*)d_in[4];
  const float* b_in1 = (const float*)d_in[5];
  const float* w_in2 = (const float*)d_in[6];
  const float* b_in2 = (const float*)d_in[7];
  const float* W     = (const float*)d_in[8];
  const float* a     = (const float*)d_in[9];
  const float* w_o1  = (const float*)d_in[10];
  const float* b_o1  = (const float*)d_in[11];
  const float* w_o2  = (const float*)d_in[12];
  const float* b_o2  = (const float*)d_in[13];
  const float* w_o3  = (const float*)d_in[14];
  const float* b_o3  = (const float*)d_in[15];
  const float* w_fc1 = (const float*)d_in[16];
  const float* b_fc1 = (const float*)d_in[17];
  const float* w_ih  = (const float*)d_in[18];
  const float* w_hh  = (const float*)d_in[19];
  const float* b_ih  = (const float*)d_in[20];
  const float* b_hh  = (const float*)d_in[21];
  const float* w_fc2 = (const float*)d_in[22];
  const float* b_fc2 = (const float*)d_in[23];

  float* ws   = (float*)d_ws;
  float* outQ = (float*)d_out;                          // B x 5
  float* outH = (float*)d_out + (size_t)BTOT * NACT;    // B x 64

  const bool packed = (ws_size >= (size_t)WS_FLOATS * sizeof(float));

  atom_rnn_prep_u<<<1, 64, 0, stream>>>(W, a, ws + OFF_U, ws + OFF_UP,
                                        packed ? 1 : 0);
  if (packed) {
    atom_rnn_pack<<<2,  256, 0, stream>>>(w_in0, ws + OFF_IN0,  8,  8,  64, 0);
    atom_rnn_pack<<<2,  256, 0, stream>>>(w_in1, ws + OFF_IN1,  8,  8,  64, 0);
    atom_rnn_pack<<<2,  256, 0, stream>>>(w_in2, ws + OFF_IN2,  8,  8,  64, 0);
    atom_rnn_pack<<<2,  256, 0, stream>>>(w_o1,  ws + OFF_O1,  10, 12,  32, 0);
    atom_rnn_pack<<<2,  256, 0, stream>>>(w_o2,  ws + OFF_O2,  32, 32,  16, 0);
    atom_rnn_pack<<<23, 256, 0, stream>>>(w_fc1, ws + OFF_FC1, 90, 92,  64, 0);
    atom_rnn_pack<<<48, 256, 0, stream>>>(w_ih,  ws + OFF_IH,  64, 64, 192, 1);
    atom_rnn_pack<<<48, 256, 0, stream>>>(w_hh,  ws + OFF_HH,  64, 64, 192, 1);
  }

  const int blocks = BTOT / (SPW * NWAVES);   // 2048
  if (packed) {
    atom_rnn_fused<true><<<blocks, NWAVES * 32, 0, stream>>>(
        obs, hid, w_in0, b_in0, w_in1, b_in1, w_in2, b_in2, ws,
        w_o1, b_o1, w_o2, b_o2, w_o3, b_o3, w_fc1, b_fc1,
        w_ih, w_hh, b_ih, b_hh, w_fc2, b_fc2, outQ, outH);
  } else {
    atom_rnn_fused<false><<<blocks, NWAVES * 32, 0, stream>>>(
        obs, hid, w_in0, b_in0, w_in1, b_in1, w_in2, b_in2, ws,
        w_o1, b_o1, w_o2, b_o2, w_o3, b_o3, w_fc1, b_fc1,
        w_ih, w_hh, b_ih, b_hh, w_fc2, b_fc2, outQ, outH);
  }
}